// VectorQuantizer_30940944401113
// MI455X (gfx1250) — compile-verified
//
#include <hip/hip_runtime.h>
#include <hip/hip_bf16.h>

// ---------------- problem constants ----------------
#define BB    16
#define DIM   256
#define HH    32
#define WW    32
#define VOCAB 8192
#define RLEV  6

typedef float v2f __attribute__((ext_vector_type(2)));
typedef float v8f __attribute__((ext_vector_type(8)));

// ---------------- output layout (floats, concat in reference return order) ---
static const long OFF_FHAT   = 0;                       // [16,256,32,32]
static const long OFF_R[6]   = {4194304, 4198400, 4214784, 4280320, 4542464, 5591040};
static const long OFF_IDX[6] = {9785344, 9785360, 9785424, 9785680, 9786704, 9790800};
static const long OFF_SCALES = 9807184;                 // [16,1364,256]
static const long OFF_LOSS   = 15394128;                // scalar

// ---------------- workspace layout (floats) ----------------
static const long WS_RESID = 0;          // 4194304
static const long WS_RF    = 4194304;    // up to 16384x256
static const long WS_T1    = 8388608;    // up to [16,256,32,32]
static const long WS_PRE   = 12582912;   // [16,256,32,32]
static const long WS_BLEND = 16777216;   // [16,256,32,32]
static const long WS_CBSQ  = 20971520;   // 8192
static const long WS_IDX   = 20979712;   // up to 16384 ints
static const long WS_PART  = 20996096;   // 6*16384 loss partials
static const long WS_WT    = 21094400;   // 4*9*256*256 transposed weights

// ---------------- gfx1250 async-LDS availability ---------------------------
#if __has_builtin(__builtin_amdgcn_global_load_async_to_lds_b32)
#define HAVE_ASYNC_LDS 1
typedef __attribute__((address_space(1))) int gas_int;
typedef __attribute__((address_space(3))) int las_int;
#else
#define HAVE_ASYNC_LDS 0
#endif

__device__ __forceinline__ void wait_asynccnt0() {
#if __has_builtin(__builtin_amdgcn_s_wait_asynccnt)
  __builtin_amdgcn_s_wait_asynccnt(0);
#else
  asm volatile("s_wait_asynccnt 0x0" ::: "memory");
#endif
}

// ---------------- bicubic taps (a = -0.75, clamped borders) ----------------
__device__ __forceinline__ void cubic_taps(int i, int n_in, int n_out,
                                           int src[4], float wt[4]) {
  const float a = -0.75f;
  float x  = (i + 0.5f) * (float)n_in / (float)n_out - 0.5f;
  float x0f = floorf(x);
  int   x0  = (int)x0f;
  float t   = x - x0f;
#pragma unroll
  for (int j = -1; j <= 2; ++j) {
    float d = fabsf(t - (float)j);
    float w;
    if (d <= 1.0f)      w = (a + 2.0f) * d * d * d - (a + 3.0f) * d * d + 1.0f;
    else if (d < 2.0f)  w = a * d * d * d - 5.0f * a * d * d + 8.0f * a * d - 4.0f * a;
    else                w = 0.0f;
    int s = x0 + j;
    s = s < 0 ? 0 : (s > n_in - 1 ? n_in - 1 : s);
    src[j + 1] = s;
    wt [j + 1] = w;
  }
}

// ---------------- codebook squared norms ----------------
__global__ void k_cbsq(const float* __restrict__ cb, float* __restrict__ cbsq) {
  int v = blockIdx.x * blockDim.x + threadIdx.x;
  if (v >= VOCAB) return;
  const float* row = cb + (size_t)v * DIM;
  float s = 0.f;
  for (int c = 0; c < DIM; ++c) s += row[c] * row[c];
  cbsq[v] = s;
}

// ---------------- weight pre-transpose: OIHW(tap-last) -> [tap][cout][cin] --
__global__ void k_wtrans(const float* __restrict__ phi_w, float* __restrict__ wt) {
  int g = blockIdx.x * blockDim.x + threadIdx.x;   // 4*9*256*256 exact
  int ci  = g % DIM;
  int co  = (g / DIM) % DIM;
  int tap = (g / (DIM * DIM)) % 9;
  int p   = g / (DIM * DIM * 9);
  wt[g] = phi_w[(((size_t)p * DIM + co) * DIM + ci) * 9 + tap];
}

// ---------------- area pool: resid -> r (BCHW out) + rf ([B*k*k, 256]) ------
__global__ void k_pool(const float* __restrict__ src, float* __restrict__ r_bchw,
                       float* __restrict__ rf, int k) {
  int g = blockIdx.x * blockDim.x + threadIdx.x;
  int total = BB * DIM * k * k;
  if (g >= total) return;
  int x = g % k;
  int y = (g / k) % k;
  int c = (g / (k * k)) % DIM;
  int b = g / (k * k * DIM);
  int bs = WW / k;
  const float* base = src + (((size_t)b * DIM + c) * HH + (size_t)y * bs) * WW + (size_t)x * bs;
  float s = 0.f;
  for (int dy = 0; dy < bs; ++dy)
    for (int dx = 0; dx < bs; ++dx) s += base[dy * WW + dx];
  s *= 1.0f / (float)(bs * bs);
  r_bchw[(((size_t)b * DIM + c) * k + y) * k + x] = s;
  rf[(((size_t)b * k + y) * k + x) * DIM + c]     = s;
}

// ---------------- WMMA argmin over codebook -------------------------------
// One wave per 16-row tile. A (16x256 f32) preloaded to VGPRs; 512 column
// tiles x 64 chained v_wmma_f32_16x16x4_f32; score = cb_sq[n] - 2*dot.
__global__ __launch_bounds__(32) void k_argmin(const float* __restrict__ rf,
                                               const float* __restrict__ cb,
                                               const float* __restrict__ cbsq,
                                               int* __restrict__ idx_int,
                                               float* __restrict__ idx_f) {
  int tile = blockIdx.x;
  int lane = threadIdx.x;
  int row  = lane & 15;         // A: M row / B: N col / C: N col
  int kp   = (lane >> 4) << 1;  // K sub-pair select {0,1} vs {2,3}

  v2f areg[64];
  const float* arow = rf + ((size_t)tile * 16 + row) * DIM + kp;
#pragma unroll
  for (int kk = 0; kk < 64; ++kk) areg[kk] = *(const v2f*)(arow + kk * 4);

  float minv[8];
  int   mini[8];
#pragma unroll
  for (int r = 0; r < 8; ++r) { minv[r] = 3.4e38f; mini[r] = 0; }

  for (int ct = 0; ct < VOCAB / 16; ++ct) {
    int col = ct * 16 + row;
    float cq = cbsq[col];
    const float* brow = cb + (size_t)col * DIM + kp;
    v8f acc = {};
#pragma unroll
    for (int kk = 0; kk < 64; ++kk) {
      v2f bfr = *(const v2f*)(brow + kk * 4);
      acc = __builtin_amdgcn_wmma_f32_16x16x4_f32(false, areg[kk], false, bfr,
                                                  (short)0, acc, false, false);
    }
#pragma unroll
    for (int r = 0; r < 8; ++r) {
      float sc = cq - 2.0f * acc[r];
      if (sc < minv[r]) { minv[r] = sc; mini[r] = col; }
    }
  }

  // reduce across the 16 lanes holding the same M (halves stay separate)
#pragma unroll
  for (int r = 0; r < 8; ++r) {
    float v = minv[r];
    int   ix = mini[r];
    for (int off = 1; off < 16; off <<= 1) {
      float ov = __shfl_xor(v, off, 32);
      int   oi = __shfl_xor(ix, off, 32);
      if (ov < v || (ov == v && oi < ix)) { v = ov; ix = oi; }
    }
    minv[r] = v; mini[r] = ix;
  }
  if (row == 0) {  // lanes 0 (M=0..7) and 16 (M=8..15)
    int mbase = (lane >> 4) * 8;
#pragma unroll
    for (int r = 0; r < 8; ++r) {
      int vec = tile * 16 + mbase + r;
      idx_int[vec] = mini[r];
      idx_f[vec]   = (float)mini[r];
    }
  }
}

// ---------------- bicubic upsample pass 1 (gather codebook fused) ----------
__global__ void k_up1(const int* __restrict__ idx_int, const float* __restrict__ cb,
                      float* __restrict__ t1, int k) {
  int g = blockIdx.x * blockDim.x + threadIdx.x;
  int total = BB * DIM * k * WW;
  if (g >= total) return;
  int W = g % WW;
  int h = (g / WW) % k;
  int c = (g / (WW * k)) % DIM;
  int b = g / (WW * k * DIM);
  int src[4]; float wt[4];
  cubic_taps(W, k, WW, src, wt);
  float s = 0.f;
#pragma unroll
  for (int j = 0; j < 4; ++j) {
    int id = idx_int[((size_t)b * k + h) * k + src[j]];
    s += wt[j] * cb[(size_t)id * DIM + c];
  }
  t1[(((size_t)b * DIM + c) * k + h) * WW + W] = s;
}

// ---------------- bicubic upsample pass 2 ----------------------------------
__global__ void k_up2(const float* __restrict__ t1, float* __restrict__ pre, int k) {
  int g = blockIdx.x * blockDim.x + threadIdx.x;  // grid exact: B*DIM*HH*WW
  int W = g % WW;
  int H = (g / WW) % HH;
  int c = (g / (WW * HH)) % DIM;
  int b = g / (WW * HH * DIM);
  int src[4]; float wt[4];
  cubic_taps(H, k, HH, src, wt);
  float s = 0.f;
#pragma unroll
  for (int j = 0; j < 4; ++j)
    s += wt[j] * t1[(((size_t)b * DIM + c) * k + src[j]) * WW + W];
  pre[g] = s;
}

// ---------------- 3x3 conv (implicit GEMM, WMMA) + 0.5/0.5 blend -----------
// Block = 8 waves. Async-LDS staged zero-padded 3x18x256 input patch; each
// wave computes two 16-pixel x 16-cout tiles with dual accumulators so every
// A fragment (ds_load_b64) feeds two v_wmma_f32_16x16x4_f32.
#define CSTRIDE 258  // 256 + 2 pad floats to dodge bank conflicts
__global__ __launch_bounds__(256) void k_conv(const float* __restrict__ pre,
                                              const float* __restrict__ wt,
                                              const float* __restrict__ phi_b,
                                              float* __restrict__ blend, int phiIdx) {
  __shared__ float patch[54 * CSTRIDE];  // [3 rows][18 cols][256 ch + pad]
  int blk = blockIdx.x;                  // B * 32 * 2
  int wt_ = blk & 1;
  int h   = (blk >> 1) & 31;
  int b   = blk >> 6;
  int w0  = wt_ * 16;
  int tid = threadIdx.x;

  // stage zero-padded 3x18x256 input patch
  for (int i = tid; i < 54 * DIM; i += 256) {
    int cell = i % 54;
    int ch   = i / 54;
    int rr = cell / 18, cc = cell % 18;
    int gh = h + rr - 1, gw = w0 + cc - 1;
    float* lp = &patch[cell * CSTRIDE + ch];
    bool valid = (gh >= 0 && gh < HH && gw >= 0 && gw < WW);
#if HAVE_ASYNC_LDS
    if (valid) {
      const float* gp = pre + (((size_t)b * DIM + ch) * HH + gh) * WW + gw;
      __builtin_amdgcn_global_load_async_to_lds_b32((gas_int*)gp, (las_int*)lp, 0, 0);
    } else {
      *lp = 0.f;
    }
#else
    float v = 0.f;
    if (valid) v = pre[(((size_t)b * DIM + ch) * HH + gh) * WW + gw];
    *lp = v;
#endif
  }
#if HAVE_ASYNC_LDS
  wait_asynccnt0();
#endif
  __syncthreads();

  int wave = tid >> 5, lane = tid & 31;
  int m     = lane & 15;          // pixel within 16-wide M tile / cout col
  int choff = (lane >> 4) << 1;   // K sub-pair select
  const float* wtp = wt + (size_t)phiIdx * 9 * DIM * DIM;  // [tap][cout][cin]
  int n0 = wave * 32 + m;
  int n1 = wave * 32 + 16 + m;

  v8f acc0 = {}, acc1 = {};
  for (int dy = 0; dy < 3; ++dy)
    for (int dx = 0; dx < 3; ++dx) {
      int tap = dy * 3 + dx;
      const float* lsrc = &patch[(dy * 18 + m + dx) * CSTRIDE + choff];
      const float* w0p  = wtp + ((size_t)tap * DIM + n0) * DIM + choff;
      const float* w1p  = wtp + ((size_t)tap * DIM + n1) * DIM + choff;
#pragma unroll 8
      for (int k0 = 0; k0 < DIM; k0 += 4) {
        v2f a  = *(const v2f*)(lsrc + k0);
        v2f b0 = *(const v2f*)(w0p + k0);
        v2f b1 = *(const v2f*)(w1p + k0);
        acc0 = __builtin_amdgcn_wmma_f32_16x16x4_f32(false, a, false, b0,
                                                     (short)0, acc0, false, false);
        acc1 = __builtin_amdgcn_wmma_f32_16x16x4_f32(false, a, false, b1,
                                                     (short)0, acc1, false, false);
      }
    }

  float bias0 = phi_b[phiIdx * DIM + n0];
  float bias1 = phi_b[phiIdx * DIM + n1];
#pragma unroll
  for (int r = 0; r < 8; ++r) {
    int mm = r + ((lane >> 4) << 3);
    size_t o0 = (((size_t)b * DIM + n0) * HH + h) * WW + (w0 + mm);
    size_t o1 = (((size_t)b * DIM + n1) * HH + h) * WW + (w0 + mm);
    blend[o0] = 0.5f * pre[o0] + 0.5f * (acc0[r] + bias0);
    blend[o1] = 0.5f * pre[o1] + 0.5f * (acc1[r] + bias1);
  }
}

// ---------------- residual / f_hat update + loss partial -------------------
__global__ __launch_bounds__(256) void k_update(const float* __restrict__ f,
                                                const float* __restrict__ blend,
                                                float* __restrict__ resid,
                                                float* __restrict__ fhat,
                                                float* __restrict__ part,
                                                int first) {
  __shared__ float red[8];
  int g = blockIdx.x * 256 + threadIdx.x;  // grid exact: 4194304
  float bl = blend[g];
  float rv = (first ? f[g] : resid[g]) - bl;
  resid[g] = rv;
  float fh = (first ? 0.f : fhat[g]) + bl;
  fhat[g] = fh;
  float d = fh - f[g];
  float d2 = d * d;
  for (int off = 16; off > 0; off >>= 1) d2 += __shfl_down(d2, off, 32);
  int lane = threadIdx.x & 31, wave = threadIdx.x >> 5;
  if (lane == 0) red[wave] = d2;
  __syncthreads();
  if (threadIdx.x == 0) {
    float s = 0.f;
    for (int i = 0; i < 8; ++i) s += red[i];
    part[blockIdx.x] = s;
  }
}

// ---------------- deterministic loss reduction -----------------------------
__global__ __launch_bounds__(256) void k_loss(const float* __restrict__ part,
                                              float* __restrict__ loss,
                                              int n, float lscale) {
  __shared__ float red[256];
  float s = 0.f;
  for (int i = threadIdx.x; i < n; i += 256) s += part[i];
  red[threadIdx.x] = s;
  __syncthreads();
  for (int off = 128; off > 0; off >>= 1) {
    if (threadIdx.x < off) red[threadIdx.x] += red[threadIdx.x + off];
    __syncthreads();
  }
  if (threadIdx.x == 0) *loss = red[0] * lscale;
}

// ---------------- scales (area pool of f_hat) ------------------------------
__global__ void k_scales(const float* __restrict__ fhat, float* __restrict__ scales,
                         int nk, int lbase) {
  int g = blockIdx.x * blockDim.x + threadIdx.x;
  int total = BB * DIM * nk * nk;
  if (g >= total) return;
  int x = g % nk;
  int y = (g / nk) % nk;
  int c = (g / (nk * nk)) % DIM;
  int b = g / (nk * nk * DIM);
  int bs = WW / nk;
  const float* base = fhat + (((size_t)b * DIM + c) * HH + (size_t)y * bs) * WW + (size_t)x * bs;
  float s = 0.f;
  for (int dy = 0; dy < bs; ++dy)
    for (int dx = 0; dx < bs; ++dx) s += base[dy * WW + dx];
  s *= 1.0f / (float)(bs * bs);
  scales[((size_t)b * 1364 + lbase + y * nk + x) * DIM + c] = s;
}

// ---------------- host driver ----------------------------------------------
extern "C" void kernel_launch(void* const* d_in, const int* in_sizes, int n_in,
                              void* d_out, int out_size, void* d_ws, size_t ws_size,
                              hipStream_t stream) {
  (void)in_sizes; (void)n_in; (void)out_size; (void)ws_size;
  const float* f     = (const float*)d_in[0];
  const float* cb    = (const float*)d_in[1];
  const float* phi_w = (const float*)d_in[2];
  const float* phi_b = (const float*)d_in[3];
  float* out = (float*)d_out;
  float* ws  = (float*)d_ws;

  float* fhat  = out + OFF_FHAT;
  float* loss  = out + OFF_LOSS;
  float* resid = ws + WS_RESID;
  float* rf    = ws + WS_RF;
  float* t1    = ws + WS_T1;
  float* pre   = ws + WS_PRE;
  float* blend = ws + WS_BLEND;
  float* cbsq  = ws + WS_CBSQ;
  int*   idxi  = (int*)(ws + WS_IDX);
  float* part  = ws + WS_PART;
  float* wtr   = ws + WS_WT;

  static const int patch[RLEV]  = {1, 2, 4, 8, 16, 32};
  static const int phidx[RLEV]  = {0, 0, 1, 2, 3, 3};
  static const int lbases[5]    = {0, 4, 20, 84, 340};

  k_cbsq<<<VOCAB / 256, 256, 0, stream>>>(cb, cbsq);
  k_wtrans<<<(4 * 9 * DIM * DIM) / 256, 256, 0, stream>>>(phi_w, wtr);

  for (int i = 0; i < RLEV; ++i) {
    int k = patch[i];
    int tot = BB * DIM * k * k;
    k_pool<<<(tot + 255) / 256, 256, 0, stream>>>(i == 0 ? f : resid,
                                                  out + OFF_R[i], rf, k);
    int ntiles = BB * k * k / 16;
    k_argmin<<<ntiles, 32, 0, stream>>>(rf, cb, cbsq, idxi, out + OFF_IDX[i]);

    int t1tot = BB * DIM * k * WW;
    k_up1<<<(t1tot + 255) / 256, 256, 0, stream>>>(idxi, cb, t1, k);
    k_up2<<<(BB * DIM * HH * WW) / 256, 256, 0, stream>>>(t1, pre, k);

    k_conv<<<BB * HH * 2, 256, 0, stream>>>(pre, wtr, phi_b, blend, phidx[i]);

    k_update<<<(BB * DIM * HH * WW) / 256, 256, 0, stream>>>(
        f, blend, resid, fhat, part + (size_t)i * 16384, i == 0 ? 1 : 0);

    if (i < RLEV - 1) {
      int nk = patch[i + 1];
      int st = BB * DIM * nk * nk;
      k_scales<<<(st + 255) / 256, 256, 0, stream>>>(fhat, out + OFF_SCALES,
                                                     nk, lbases[i]);
    }
  }
  k_loss<<<1, 256, 0, stream>>>(part, loss, RLEV * 16384,
                                1.25f / ((float)RLEV * 4194304.0f));
}